// DirDist_P2P_9723805958691
// MI455X (gfx1250) — compile-verified
//
#include <hip/hip_runtime.h>
#include <math.h>

typedef __attribute__((ext_vector_type(2))) float v2f;
typedef __attribute__((ext_vector_type(8))) float v8f;

#define NPTS   2048
#define UPR    10
#define QTOT   (NPTS * UPR + NPTS)   /* 22528 */
#define NTILES (NPTS / 16)           /* 128 */
#define QTILES (QTOT / 16)           /* 1408 */
#define KNN    5
#define PSTD   0.05f
#define BETA   3.0f
#define REDBLK 256
#define NPART  (QTOT / REDBLK)       /* 88 */

// ---------------------------------------------------------------------------
// Branchless sorted top-5 insert (ascending). Compiles to v_cmp + v_cndmask,
// no EXEC manipulation -> WMMA's EXEC==all-ones requirement holds trivially.
// Update order bottom-up so each level reads the OLD value of the level above.
// ---------------------------------------------------------------------------
__device__ __forceinline__ void ins5(float (&d)[KNN], int (&ix)[KNN],
                                     float v, int n) {
  const bool c0 = v < d[0];
  const bool c1 = v < d[1];
  const bool c2 = v < d[2];
  const bool c3 = v < d[3];
  const bool c4 = v < d[4];
  d[4]  = c4 ? (c3 ? d[3]  : v) : d[4];
  ix[4] = c4 ? (c3 ? ix[3] : n) : ix[4];
  d[3]  = c3 ? (c2 ? d[2]  : v) : d[3];
  ix[3] = c3 ? (c2 ? ix[2] : n) : ix[3];
  d[2]  = c2 ? (c1 ? d[1]  : v) : d[2];
  ix[2] = c2 ? (c1 ? ix[1] : n) : ix[2];
  d[1]  = c1 ? (c0 ? d[0]  : v) : d[1];
  ix[1] = c1 ? (c0 ? ix[0] : n) : ix[1];
  d[0]  = c0 ? v : d[0];
  ix[0] = c0 ? n : ix[0];
}

// ---------------------------------------------------------------------------
// Kernel A: build jittered query points, pad with ||q||^2 in .w
// ---------------------------------------------------------------------------
__global__ void k_build_query(const float* __restrict__ src,
                              const float* __restrict__ tgt,
                              const float* __restrict__ noise,
                              float4* __restrict__ q4) {
  int q = blockIdx.x * blockDim.x + threadIdx.x;
  if (q >= QTOT) return;
  float x, y, z;
  if (q < NPTS * UPR) {
    int m = q / UPR;
    x = tgt[m * 3 + 0] + noise[q * 3 + 0] * PSTD;
    y = tgt[m * 3 + 1] + noise[q * 3 + 1] * PSTD;
    z = tgt[m * 3 + 2] + noise[q * 3 + 2] * PSTD;
  } else {
    int s = q - NPTS * UPR;
    x = src[s * 3 + 0];
    y = src[s * 3 + 1];
    z = src[s * 3 + 2];
  }
  q4[q] = make_float4(x, y, z, x * x + y * y + z * z);
}

// ---------------------------------------------------------------------------
// Kernel B: augmented cloud descriptors (-2x, -2y, -2z, ||x||^2)
// so that  (qx,qy,qz,1) . Bcol = ||x||^2 - 2 q.x   (add ||q||^2 later)
// ---------------------------------------------------------------------------
__global__ void k_build_baug(const float* __restrict__ src,
                             const float* __restrict__ tgt,
                             float4* __restrict__ b0,
                             float4* __restrict__ b1) {
  int i = blockIdx.x * blockDim.x + threadIdx.x;
  if (i >= 2 * NPTS) return;
  const float* p = (i < NPTS) ? tgt : src;
  float4* o = (i < NPTS) ? b0 : b1;
  int n = (i < NPTS) ? i : i - NPTS;
  float x = p[n * 3 + 0], y = p[n * 3 + 1], z = p[n * 3 + 2];
  o[n] = make_float4(-2.f * x, -2.f * y, -2.f * z, x * x + y * y + z * z);
}

// ---------------------------------------------------------------------------
// Kernel C: one wave32 per 16-query tile, grid.y selects cloud (0=tgt,1=src).
// Distance tiles via V_WMMA_F32_16X16X4_F32; branchless per-lane top-5,
// LDS merge, then weights / udf_grad / udf per query row.
// ---------------------------------------------------------------------------
__launch_bounds__(32)
__global__ void k_knn_udf(const float4* __restrict__ q4,
                          const float4* __restrict__ baug0,
                          const float4* __restrict__ baug1,
                          float4* __restrict__ res) {
  __shared__ float4 sB[NPTS];              // 32 KB: full augmented cloud
  __shared__ float  sD[16][16][KNN];       // per-row / per-lane-slot candidates
  __shared__ int    sI[16][16][KNN];

  const int lane  = threadIdx.x;           // 0..31 (wave32)
  const int tile  = blockIdx.x;            // query tile
  const int cloud = blockIdx.y;            // 0 = tgt, 1 = src
  const float4* baug = cloud ? baug1 : baug0;

  for (int i = lane; i < NPTS; i += 32) sB[i] = baug[i];
  __syncthreads();

  // A fragment (16x4, f32): lanes 0-15 hold K=0,1 ; lanes 16-31 hold K=2,3
  const int rlo = lane & 15;
  const int hi  = lane >> 4;
  float4 qq = q4[tile * 16 + rlo];
  v2f a;
  a[0] = hi ? qq.z : qq.x;
  a[1] = hi ? 1.0f : qq.y;

  // per-lane sorted top-5 for the 8 C-matrix rows this lane holds
  float td[8][KNN];
  int   ti[8][KNN];
#pragma unroll
  for (int v = 0; v < 8; ++v)
#pragma unroll
    for (int j = 0; j < KNN; ++j) { td[v][j] = 3.0e38f; ti[v][j] = 0; }

  for (int t = 0; t < NTILES; ++t) {
    int n = t * 16 + rlo;                  // this lane's column index
    float4 bb = sB[n];                     // ds_load_b128
    v2f b;                                 // B 4x16 mirrors A's K split
    b[0] = hi ? bb.z : bb.x;               // K=0 / K=2
    b[1] = hi ? bb.w : bb.y;               // K=1 / K=3
    v8f c = {};
    v8f d = __builtin_amdgcn_wmma_f32_16x16x4_f32(
        false, a, false, b, (short)0, c, false, false);
#pragma unroll
    for (int v = 0; v < 8; ++v)            // row (v + 8*hi), column n
      ins5(td[v], ti[v], d[v], n);
  }

  // dump per-lane candidate lists
#pragma unroll
  for (int v = 0; v < 8; ++v) {
    int row = v + 8 * hi;
#pragma unroll
    for (int j = 0; j < KNN; ++j) {
      sD[row][rlo][j] = td[v][j];
      sI[row][rlo][j] = ti[v][j];
    }
  }
  __syncthreads();

  // merge 16 lists per row; lanes 0..15 each own one query row
  if (lane < 16) {
    int row = lane;
    float bd[KNN]; int bi[KNN];
#pragma unroll
    for (int j = 0; j < KNN; ++j) { bd[j] = sD[row][0][j]; bi[j] = sI[row][0][j]; }
    for (int s = 1; s < 16; ++s)
#pragma unroll
      for (int j = 0; j < KNN; ++j)
        ins5(bd, bi, sD[row][s][j], sI[row][s][j]);

    int q = tile * 16 + row;
    float4 qp = q4[q];                     // xyz + ||q||^2 in .w
    float inv[KNN]; float sinv = 0.f;
#pragma unroll
    for (int j = 0; j < KNN; ++j) {
      float dist = bd[j] + qp.w;           // complete squared distance
      dist = dist > 0.f ? dist : 0.f;      // clamp fp roundoff
      inv[j] = 1.0f / (dist + 1e-8f);
      sinv += inv[j];
    }
    float gx = 0.f, gy = 0.f, gz = 0.f;
#pragma unroll
    for (int j = 0; j < KNN; ++j) {
      float w = inv[j] / sinv;
      float4 bb = sB[bi[j]];               // recover point: x = -0.5 * (-2x)
      gx += (qp.x + 0.5f * bb.x) * w;
      gy += (qp.y + 0.5f * bb.y) * w;
      gz += (qp.z + 0.5f * bb.z) * w;
    }
    float ux = gx + 1e-10f, uy = gy + 1e-10f, uz = gz + 1e-10f;
    float udf = sqrtf(ux * ux + uy * uy + uz * uz);
    res[cloud * QTOT + q] = make_float4(udf, gx, gy, gz);
  }
}

// ---------------------------------------------------------------------------
// Kernel D: per-query error + deterministic block reduction
// ---------------------------------------------------------------------------
__global__ void k_err_partial(const float4* __restrict__ res,
                              float* __restrict__ part) {
  __shared__ float sred[REDBLK];
  int q = blockIdx.x * REDBLK + threadIdx.x;
  float c = 0.f;
  if (q < QTOT) {
    float4 rt = res[q];
    float4 rs = res[QTOT + q];
    float err = fabsf(rt.x - rs.x);
    float ger = fabsf(rs.y - rt.y) + fabsf(rs.z - rt.z) + fabsf(rs.w - rt.w);
    float e = err + ger;
    c = e * expf(-BETA * e);               // exp(-err*B)*exp(-gerr*B)
  }
  sred[threadIdx.x] = c;
  __syncthreads();
  for (int s = REDBLK / 2; s > 0; s >>= 1) {
    if (threadIdx.x < s) sred[threadIdx.x] += sred[threadIdx.x + s];
    __syncthreads();
  }
  if (threadIdx.x == 0) part[blockIdx.x] = sred[0];
}

// Kernel E: fixed-order final sum (deterministic, no float atomics)
__global__ void k_final(const float* __restrict__ part, float* __restrict__ out) {
  if (threadIdx.x == 0 && blockIdx.x == 0) {
    float s = 0.f;
    for (int i = 0; i < NPART; ++i) s += part[i];
    out[0] = s / (float)QTOT;              // / B(=1) / Q
  }
}

// ---------------------------------------------------------------------------
extern "C" void kernel_launch(void* const* d_in, const int* in_sizes, int n_in,
                              void* d_out, int out_size, void* d_ws, size_t ws_size,
                              hipStream_t stream) {
  (void)in_sizes; (void)n_in; (void)out_size; (void)ws_size;
  const float* src   = (const float*)d_in[0];   // [1,2048,3]
  const float* tgt   = (const float*)d_in[1];   // [1,2048,3]
  const float* noise = (const float*)d_in[2];   // [1,2048,10,3]
  float* out = (float*)d_out;

  float4* q4   = (float4*)d_ws;                 // QTOT float4
  float4* b0   = q4 + QTOT;                     // NPTS float4 (tgt aug)
  float4* b1   = b0 + NPTS;                     // NPTS float4 (src aug)
  float4* res  = b1 + NPTS;                     // 2*QTOT float4
  float*  part = (float*)(res + 2 * QTOT);      // NPART floats

  k_build_query<<<(QTOT + 255) / 256, 256, 0, stream>>>(src, tgt, noise, q4);
  k_build_baug<<<(2 * NPTS + 255) / 256, 256, 0, stream>>>(src, tgt, b0, b1);
  k_knn_udf<<<dim3(QTILES, 2), 32, 0, stream>>>(q4, b0, b1, res);
  k_err_partial<<<NPART, REDBLK, 0, stream>>>(res, part);
  k_final<<<1, 1, 0, stream>>>(part, out);
}